// Attention_58488864637404
// MI455X (gfx1250) — compile-verified
//
#include <hip/hip_runtime.h>

#define B_  32
#define T_  4096
#define H_  512
#define KW  11
#define PADW 5
#define NCHUNK 16
#define CT  (T_ / NCHUNK)          // 256 timesteps per workgroup
#define TT  32                     // timesteps per LDS tile (32*512*4 = 64 KB)
#define NTILE (CT / TT)            // 8 tiles, double-buffered
#define NTHR 256                   // 8 waves of 32
#define NASY (TT * H_ / 4 / NTHR)  // 16 async b128 issues per thread per tile

#if defined(__has_builtin)
# if __has_builtin(__builtin_amdgcn_global_load_async_to_lds_b128)
#  define USE_ASYNC_LDS 1
# endif
# if __has_builtin(__builtin_amdgcn_wmma_f32_16x16x4_f32)
#  define USE_WMMA 1
# endif
#endif

typedef float v2f __attribute__((ext_vector_type(2)));
typedef float v8f __attribute__((ext_vector_type(8)));

#if defined(USE_ASYNC_LDS)
// Param 0 confirmed by hipcc diagnostic: 'int4 __device__ *' (AS1 v4i32*).
typedef int v4i_ __attribute__((ext_vector_type(4)));
typedef __attribute__((address_space(1))) v4i_* as1_v4i_p;
typedef __attribute__((address_space(3))) v4i_* as3_v4i_p;

# if __has_builtin(__builtin_amdgcn_s_wait_asynccnt)
#  define WAIT_ASYNC(n) __builtin_amdgcn_s_wait_asynccnt(n)
# else
#  define WAIT_ASYNC(n) asm volatile("s_wait_asynccnt %0" :: "i"(n) : "memory")
# endif

__device__ __forceinline__ void issue_tile_async(const float4* gsrc, float4* lbuf,
                                                 int tid)
{
#pragma unroll
    for (int k = 0; k < NASY; ++k) {
        const int i = tid + k * NTHR;
        __builtin_amdgcn_global_load_async_to_lds_b128(
            (as1_v4i_p)(float4*)(gsrc + i), (as3_v4i_p)(lbuf + i), 0, 0);
    }
}
#else
# define WAIT_ASYNC(n) ((void)0)
__device__ __forceinline__ void issue_tile_async(const float4* gsrc, float4* lbuf,
                                                 int tid)
{
#pragma unroll
    for (int k = 0; k < NASY; ++k) {
        const int i = tid + k * NTHR;
        lbuf[i] = gsrc[i];
    }
}
#endif

// ---------------------------------------------------------------------------
// Pass 1: flash-style fused score + online-softmax + weighted context partials.
// One workgroup per (batch, T-chunk). Reads eh exactly ONCE from HBM, with
// double-buffered async global->LDS copies overlapping compute. Context
// accumulation runs on the WMMA unit (v_wmma_f32_16x16x4_f32): A = softmax
// weights broadcast across all 16 rows, B = eh columns, so the k-sum is
// layout-permutation invariant and every lane reads its result from D[0].
// ---------------------------------------------------------------------------
__global__ __launch_bounds__(NTHR)
void attn_pass1(const float* __restrict__ eh, const float* __restrict__ dhx,
                const float* __restrict__ ax, const float* __restrict__ cw,
                const float* __restrict__ cb,
                float* __restrict__ pax,
                float* __restrict__ pm, float* __restrict__ pl,
                float* __restrict__ pacc)
{
    __shared__ float4 lds_buf[2][TT * H_ / 4];     // 2 x 64 KB double buffer
    __shared__ __align__(16) float lds_dhx[H_];
    __shared__ float lds_s[TT];
    __shared__ float lds_w[TT];
    __shared__ float sh_cw[KW];
    __shared__ float sh_scale, sh_runm, sh_runl;

    const int tid   = threadIdx.x;
    const int b     = blockIdx.x / NCHUNK;
    const int chunk = blockIdx.x % NCHUNK;
    const int t0    = chunk * CT;

    // straight-line staging: 256 threads x float2 = 512 floats of dhx
    reinterpret_cast<float2*>(lds_dhx)[tid] =
        reinterpret_cast<const float2*>(dhx + b * H_)[tid];
    if (tid < KW) sh_cw[tid] = cw[tid];
    if (tid == 0) { sh_runm = -__builtin_inff(); sh_runl = 0.f; }
    const float cbv = cb[0];

    const float4* gbase =
        reinterpret_cast<const float4*>(eh + ((size_t)b * T_ + t0) * H_);

    // prime the pipeline: tile 0 -> buffer 0
    issue_tile_async(gbase, &lds_buf[0][0], tid);

#if defined(USE_WMMA)
    const int wave = tid >> 5;
    const int lane = tid & 31;
    const int n16  = lane & 15;     // N position within the 16-wide h group
    const int kh   = lane >> 4;     // lane-half selects which K pair we feed
    v8f cacc[4] = {};               // 4 x 16-h groups: wave owns h = wave*64..+63
#else
    float acc0 = 0.f, acc1 = 0.f;   // running context, h = 2*tid, 2*tid+1
    const int hpair = 2 * tid;
#endif

    for (int tile = 0; tile < NTILE; ++tile) {
        const float* cur = reinterpret_cast<const float*>(&lds_buf[tile & 1][0]);

        if (tile + 1 < NTILE) {
            // prefetch next tile into the other buffer, then wait only for
            // the CURRENT tile's 16 per-wave async loads (in-order completion)
            issue_tile_async(gbase + (size_t)(tile + 1) * (TT * H_ / 4),
                             &lds_buf[(tile + 1) & 1][0], tid);
            WAIT_ASYNC(NASY);
        } else {
            WAIT_ASYNC(0);
        }
        __syncthreads();                   // tile resident, all waves synced

        // ---- content + location score: 8 lanes per timestep ----
        {
            const int g = tid >> 3, l = tid & 7;       // g: t-row 0..31
            const float4* ev =
                reinterpret_cast<const float4*>(cur + g * H_ + l * (H_ / 8));
            const float4* dv =
                reinterpret_cast<const float4*>(lds_dhx + l * (H_ / 8));
            float p = 0.f;
#pragma unroll
            for (int j = 0; j < H_ / 32; ++j) {
                const float4 e = ev[j], d = dv[j];
                p += e.x * d.x + e.y * d.y + e.z * d.z + e.w * d.w;
            }
            p += __shfl_xor(p, 1, 32);
            p += __shfl_xor(p, 2, 32);
            p += __shfl_xor(p, 4, 32);
            if (l == 0) {
                const int t = t0 + tile * TT + g;
                float loc = cbv;
#pragma unroll
                for (int k = 0; k < KW; ++k) {
                    const int ti = t - PADW + k;
                    loc += (ti >= 0 && ti < T_) ? sh_cw[k] * ax[b * T_ + ti] : 0.f;
                }
                const float s = p + loc;
                lds_s[g] = s;
                pax[(size_t)b * T_ + t] = s;   // raw score for final softmax
            }
        }
        __syncthreads();

        // ---- wave0: tile max, weights, running (m, l) update ----
        if (tid < 32) {
            const float s = lds_s[tid];
            float m = s;
#pragma unroll
            for (int off = 16; off > 0; off >>= 1)
                m = fmaxf(m, __shfl_xor(m, off, 32));
            const float rm = sh_runm;
            const float nm = fmaxf(rm, m);
            const float w = __expf(s - nm);
            lds_w[tid] = w;
            float ws = w;
#pragma unroll
            for (int off = 16; off > 0; off >>= 1)
                ws += __shfl_xor(ws, off, 32);
            if (tid == 0) {
                const float sc = __expf(rm - nm);   // exp(-inf)=0 on first tile
                sh_scale = sc;
                sh_runl  = sh_runl * sc + ws;
                sh_runm  = nm;
            }
        }
        __syncthreads();

        // ---- rescale + accumulate context ----
        const float sc = sh_scale;
#if defined(USE_WMMA)
        // fully converged code: EXEC is all-ones as WMMA requires
#pragma unroll
        for (int g = 0; g < 4; ++g) cacc[g] = cacc[g] * sc;
#pragma unroll
        for (int ks = 0; ks < TT / 4; ++ks) {      // 8 k-steps of 4 timesteps
            const int tb = ks * 4 + 2 * kh;        // even -> aligned b64 load
            const v2f a = *reinterpret_cast<const v2f*>(lds_w + tb);
#pragma unroll
            for (int g = 0; g < 4; ++g) {
                const int h = wave * 64 + g * 16 + n16;
                v2f bv;
                bv.x = cur[tb * H_ + h];
                bv.y = cur[(tb + 1) * H_ + h];
                cacc[g] = __builtin_amdgcn_wmma_f32_16x16x4_f32(
                    false, a, false, bv, (short)0, cacc[g], false, false);
            }
        }
#else
        float a0 = acc0 * sc, a1 = acc1 * sc;
#pragma unroll 8
        for (int t = 0; t < TT; ++t) {
            const float w = lds_w[t];
            const float2 e2 =
                *reinterpret_cast<const float2*>(cur + t * H_ + hpair);
            a0 += w * e2.x;
            a1 += w * e2.y;
        }
        acc0 = a0; acc1 = a1;
#endif
        __syncthreads();                   // buffer fully consumed -> reusable
    }

    const int pidx = b * NCHUNK + chunk;
    if (tid == 0) { pm[pidx] = sh_runm; pl[pidx] = sh_runl; }
#if defined(USE_WMMA)
    if (kh == 0) {                         // lanes 16-31 hold duplicate rows
#pragma unroll
        for (int g = 0; g < 4; ++g)
            pacc[(size_t)pidx * H_ + wave * 64 + g * 16 + n16] = cacc[g][0];
    }
#else
    pacc[(size_t)pidx * H_ + hpair]     = acc0;
    pacc[(size_t)pidx * H_ + hpair + 1] = acc1;
#endif
}

// ---------------------------------------------------------------------------
// Pass 2: one block per batch. Merge chunk partials -> global (M, L),
// emit sx = context / L, and normalize the alignment in place.
// ---------------------------------------------------------------------------
__global__ __launch_bounds__(NTHR)
void attn_pass2(const float* __restrict__ pm, const float* __restrict__ pl,
                const float* __restrict__ pacc,
                float* __restrict__ pax,      // in: raw scores, out: softmax a
                float* __restrict__ sx)       // [B][H]
{
    __shared__ float shM, shL;
    const int tid = threadIdx.x;
    const int b   = blockIdx.x;

    if (tid == 0) {
        float M = -__builtin_inff();
        for (int p = 0; p < NCHUNK; ++p) M = fmaxf(M, pm[b * NCHUNK + p]);
        float L = 0.f;
        for (int p = 0; p < NCHUNK; ++p)
            L += pl[b * NCHUNK + p] * __expf(pm[b * NCHUNK + p] - M);
        shM = M; shL = L;
    }
    __syncthreads();
    const float M = shM;
    const float invL = 1.0f / shL;

    float s0 = 0.f, s1 = 0.f;
    for (int p = 0; p < NCHUNK; ++p) {
        const float f = __expf(pm[b * NCHUNK + p] - M);
        const float* ap = pacc + (size_t)(b * NCHUNK + p) * H_;
        s0 += f * ap[tid];
        s1 += f * ap[tid + NTHR];
    }
    sx[b * H_ + tid]        = s0 * invL;
    sx[b * H_ + tid + NTHR] = s1 * invL;

    for (int t = tid; t < T_; t += NTHR) {
        const float v = pax[(size_t)b * T_ + t];
        pax[(size_t)b * T_ + t] = __expf(v - M) * invL;
    }
}

// ---------------------------------------------------------------------------
extern "C" void kernel_launch(void* const* d_in, const int* in_sizes, int n_in,
                              void* d_out, int out_size, void* d_ws, size_t ws_size,
                              hipStream_t stream)
{
    (void)in_sizes; (void)n_in; (void)out_size; (void)ws_size;
    const float* eh  = (const float*)d_in[0];   // (B,T,H)
    const float* dhx = (const float*)d_in[1];   // (B,H)
    const float* ax  = (const float*)d_in[2];   // (B,T)
    const float* cw  = (const float*)d_in[3];   // (1,1,K)
    const float* cb  = (const float*)d_in[4];   // (1,)

    float* out = (float*)d_out;
    float* sx  = out;                 // (B,1,H) flat = B*H floats
    float* a   = out + B_ * H_;       // (B,T) flat — holds raw scores between passes

    float* pm   = (float*)d_ws;                 // B*NCHUNK
    float* pl   = pm + B_ * NCHUNK;             // B*NCHUNK
    float* pacc = pl + B_ * NCHUNK;             // B*NCHUNK*H  (~1 MB total)

    attn_pass1<<<B_ * NCHUNK, NTHR, 0, stream>>>(eh, dhx, ax, cw, cb, a, pm, pl, pacc);
    attn_pass2<<<B_, NTHR, 0, stream>>>(pm, pl, pacc, a, sx);
}